// Matching_Reducer_26190710571399
// MI455X (gfx1250) — compile-verified
//
#include <hip/hip_runtime.h>
#include <hip/hip_bf16.h>
#include <stdint.h>

typedef float v2f __attribute__((ext_vector_type(2)));
typedef float v8f __attribute__((ext_vector_type(8)));

#define B_   32
#define H_   50
#define S_   64
#define D_   256
#define K_   5
#define BH   (B_ * H_)   /* 1600 */
#define SM1  (S_ - 1)    /* 63   */

// ---------------------------------------------------------------------------
// Kernel 1: q = [user | news] @ W_align^T + b ; qn = q / max(|q|, eps)
// One wave (32 threads) computes one 16x16 tile of q via V_WMMA_F32_16X16X4_F32.
// M = BH = 1600 (grid.x = 100), N = D = 256 (grid.y = 16), K = 2D = 512.
// A-frag (16x4 f32): lanes 0-15 hold M=0..15 with VGPR0=K0,VGPR1=K1;
//                    lanes 16-31 hold K2,K3.  B-frag mirrors with N on lanes.
// C/D (16x16 f32): VGPR v, lane l -> (M = v + 8*(l>=16), N = l%16).
// ---------------------------------------------------------------------------
__global__ __launch_bounds__(32)
void qn_wmma_kernel(const float* __restrict__ user,   // (B, 256)
                    const float* __restrict__ news,   // (BH, 256)
                    const float* __restrict__ W,      // (256, 512) row-major
                    const float* __restrict__ bias,   // (256)
                    float* __restrict__ qn)           // (BH, 256)
{
    const int lane = threadIdx.x;
    const int lo   = lane & 15;
    const int hi   = lane >> 4;
    const int m0   = blockIdx.x * 16;
    const int n0   = blockIdx.y * 16;
    const int m    = m0 + lo;           // A row this lane supplies
    const int n    = n0 + lo;           // B column this lane supplies
    const int bb   = m / H_;            // batch index of row m

    v8f c = {};

    const float* Wrow = W + (size_t)n * 512;   // W[n][k] = (W^T)[k][n]
    const float* Au   = user + (size_t)bb * 256;
    const float* An   = news + (size_t)m  * 256;

    // K in [0,256): A comes from broadcast user_repr
    for (int k0 = 0; k0 < 256; k0 += 4) {
        const int kA = k0 + 2 * hi;
        v2f a;  a.x  = Au[kA];       a.y  = Au[kA + 1];
        v2f bf; bf.x = Wrow[kA];     bf.y = Wrow[kA + 1];
        c = __builtin_amdgcn_wmma_f32_16x16x4_f32(
                false, a, false, bf, (short)0, c, false, false);
    }
    // K in [256,512): A comes from news_repr
    for (int k0 = 256; k0 < 512; k0 += 4) {
        const int kA = k0 + 2 * hi;
        v2f a;  a.x  = An[kA - 256]; a.y  = An[kA - 255];
        v2f bf; bf.x = Wrow[kA];     bf.y = Wrow[kA + 1];
        c = __builtin_amdgcn_wmma_f32_16x16x4_f32(
                false, a, false, bf, (short)0, c, false, false);
    }

    const float bn = bias[n];
#pragma unroll
    for (int v = 0; v < 8; ++v) {
        const int mm = m0 + v + 8 * hi;
        const float q = c[v] + bn;
        qn[(size_t)mm * 256 + n] = q / fmaxf(fabsf(q), 1e-12f);
    }
}

// ---------------------------------------------------------------------------
// Kernel 2: per-(b,h) block. Scores = dot(sel_row, qn)/max(||sel_row||,eps),
// masked -inf; exact top-5 with jax.lax.top_k tie-breaking; gather+scale emb.
// ---------------------------------------------------------------------------
__global__ __launch_bounds__(256)
void score_topk_gather_kernel(const float* __restrict__ sel,     // (BH,64,256)
                              const float* __restrict__ emb,     // (BH,64,256)
                              const float* __restrict__ qn,      // (BH,256)
                              const uint8_t* __restrict__ mask,  // (BH,64)
                              const uint8_t* __restrict__ maskk, // (BH,64)
                              float* __restrict__ out)
{
    __shared__ float qn_s[256];
    __shared__ float sc_s[SM1];
    __shared__ int   kid_s[K_];
    __shared__ float kval_s[K_];
    __shared__ float kscale_s[K_];

    const int bh   = blockIdx.x;
    const int t    = threadIdx.x;
    const int lane = t & 31;
    const int wave = t >> 5;

    qn_s[t] = qn[(size_t)bh * 256 + t];
    __syncthreads();

    // ---- scores for rows s = 0..62 (original row s+1), 8 waves round-robin
    const float* selbh = sel + (size_t)bh * (S_ * D_);
    for (int s = wave; s < SM1; s += 8) {
        const float* row = selbh + (size_t)(s + 1) * D_;
        float dot = 0.f, ss = 0.f;
#pragma unroll
        for (int i = 0; i < 8; ++i) {
            const float x = row[lane + 32 * i];      // coalesced
            ss  += x * x;
            dot += x * qn_s[lane + 32 * i];          // bank-conflict-free
        }
#pragma unroll
        for (int off = 16; off > 0; off >>= 1) {     // wave32 reduction
            ss  += __shfl_xor(ss,  off, 32);
            dot += __shfl_xor(dot, off, 32);
        }
        if (lane == 0) {
            float sc = dot / fmaxf(sqrtf(ss), 1e-12f);
            if (!maskk[bh * S_ + s + 1]) sc = -INFINITY;
            sc_s[s] = sc;
        }
    }
    __syncthreads();

    // ---- exact top-5, descending, lower index wins ties (jax.lax.top_k)
    if (t == 0) {
        unsigned long long used = 0ull;
        for (int j = 0; j < K_; ++j) {
            int bid = -1; float best = 0.f;
            for (int s = 0; s < SM1; ++s) {
                if (used & (1ull << s)) continue;
                const float v = sc_s[s];
                if (bid < 0 || v > best) { best = v; bid = s; }
            }
            used |= (1ull << bid);
            kid_s[j]    = bid;
            kval_s[j]   = best;
            kscale_s[j] = (best < 0.1f) ? 0.f : best;   // THRESHOLD, kills -inf too
        }
    }
    __syncthreads();

    // ---- ps_term_mask (float 0/1) and score_kid (float) outputs
    if (t < K_) {
        const int   kid = kid_s[t];
        const float val = kval_s[t];
        const bool  mp  = val < 0.1f;
        const bool  pm  = (mask[bh * S_ + kid + 1] != 0) && !mp;
        const size_t mask_off = (size_t)BH * K_ * D_;
        const size_t kid_off  = mask_off + (size_t)BH * K_;
        out[mask_off + (size_t)bh * K_ + t] = pm ? 1.0f : 0.0f;
        out[kid_off  + (size_t)bh * K_ + t] = (float)kid;
    }

    // ---- gather + scale 5 emb rows, coalesced over t
    const float* embbh = emb + (size_t)bh * (S_ * D_);
#pragma unroll
    for (int j = 0; j < K_; ++j) {
        out[((size_t)bh * K_ + j) * D_ + t] =
            embbh[(size_t)(kid_s[j] + 1) * D_ + t] * kscale_s[j];
    }
}

extern "C" void kernel_launch(void* const* d_in, const int* in_sizes, int n_in,
                              void* d_out, int out_size, void* d_ws, size_t ws_size,
                              hipStream_t stream) {
    const float*   sel   = (const float*)d_in[0];   // news_selection_embedding
    const float*   emb   = (const float*)d_in[1];   // news_embedding
    const float*   user  = (const float*)d_in[2];   // user_repr
    const float*   news  = (const float*)d_in[3];   // news_repr
    const float*   W     = (const float*)d_in[4];   // W_align
    const float*   bias  = (const float*)d_in[5];   // b_align
    const uint8_t* mask  = (const uint8_t*)d_in[6]; // his_attn_mask
    const uint8_t* maskk = (const uint8_t*)d_in[7]; // his_attn_mask_k
    float* out = (float*)d_out;
    float* qn  = (float*)d_ws;                      // 1600*256 f32 = 1.6 MB

    dim3 g1(BH / 16, D_ / 16);                      // 100 x 16 tiles
    qn_wmma_kernel<<<g1, 32, 0, stream>>>(user, news, W, bias, qn);
    score_topk_gather_kernel<<<BH, 256, 0, stream>>>(sel, emb, qn, mask, maskk, out);
}